// LocallyConnected1d_76665166233716
// MI455X (gfx1250) — compile-verified
//
#include <hip/hip_runtime.h>
#include <stdint.h>

#define USE_ASYNC_LDS 1

typedef __attribute__((ext_vector_type(2))) float v2f;
typedef __attribute__((ext_vector_type(8))) float v8f;

namespace {
constexpr int kB      = 128;   // batch
constexpr int kCin    = 64;
constexpr int kCout   = 64;
constexpr int kL      = 1024;
constexpr int kK      = 9;
constexpr int kLout   = 1016;        // L - K + 1
constexpr int kKT     = kCin * kK;   // 576 flattened (c,k) reduction dim
constexpr int kKC     = 32;          // K-chunk staged in LDS
constexpr int kChunks = kKT / kKC;   // 18 (even -> clean 2x pipeline)
constexpr int kPairs  = kKC / 2;     // 16 K-pairs per chunk
// Pair-row strides in float2 units; >= row length and == 16 (mod 32) so lanes
// 0-15 (row tp) and 16-31 (row tp+1) of a wave-wide b64 read hit disjoint
// bank halves.
constexpr int kXsStr  = 144;         // >= 128 batch rows
constexpr int kWsStr  = 80;          // >= 64 out channels
constexpr int kXsN    = kPairs * kXsStr;   // f2 elements per X buffer
constexpr int kWsN    = kPairs * kWsStr;   // f2 elements per W buffer
constexpr int kOpsPerFill = 16 + 8;  // async ops issued per thread per chunk
}

// One workgroup (8 wave32) per output position i.
// Wave (mg,ng): M rows [mg*32,+32), N cols [ng*32,+32) -> 2x2 WMMA tiles.
// Double-buffered LDS; async fills for chunk ch+1 overlap compute of chunk ch.
__global__ __launch_bounds__(256)
void lc1d_wmma(const float* __restrict__ x,     // (B, C_IN, L)
               const float* __restrict__ w,     // (C_OUT, C_IN, L_OUT, K)
               const float* __restrict__ bias,  // (C_OUT, L_OUT)
               float* __restrict__ out)         // (B, C_OUT, L_OUT)
{
    __shared__ v2f Xs2[2 * kXsN];   // [buf][tp][m]   K-pair interleaved
    __shared__ v2f Ws2[2 * kWsN];   // [buf][tp][o]

    const int i    = blockIdx.x;
    const int tid  = threadIdx.x;
    const int wave = tid >> 5;
    const int lane = tid & 31;
    const int half = lane >> 4;     // K row tp vs tp+1; M vs M+8 in C/D
    const int l16  = lane & 15;
    const int mg   = wave >> 1;     // M group (32 rows)
    const int ng   = wave & 1;      // N group (32 cols)

    v8f acc[2][2] = { { v8f{}, v8f{} }, { v8f{}, v8f{} } };

    // X fill: 16 consecutive t for one batch row (runs of 9 contiguous floats)
    const int xm  = tid >> 1;             // batch row 0..127
    const int xt0 = (tid & 1) * 16;
    // W fill: 8 consecutive t for one output channel
    const int wo  = tid >> 2;             // out channel 0..63
    const int wt0 = (tid & 3) * 8;

#if USE_ASYNC_LDS
    const uint32_t xsBase = (uint32_t)(uintptr_t)(float*)Xs2;  // LDS offset
    const uint32_t wsBase = (uint32_t)(uintptr_t)(float*)Ws2;
#endif

    // ---- issue async fills for chunk `ch` into buffer `buf` ----
    auto fill = [&](int ch, int buf) {
        const int tb = ch * kKC;
        #pragma unroll
        for (int j = 0; j < 16; ++j) {
            const int tl   = xt0 + j;
            const int t    = tb + tl;
            const int c    = t / kK;
            const int k    = t - c * kK;
            const int flat = buf * (2 * kXsN) + (tl >> 1) * (2 * kXsStr) + 2 * xm + (tl & 1);
            const uint32_t goff = (uint32_t)((((xm * kCin + c) * kL) + i + k) * 4);
#if USE_ASYNC_LDS
            asm volatile("global_load_async_to_lds_b32 %0, %1, %2"
                         :: "v"(xsBase + (uint32_t)flat * 4u), "v"(goff), "s"(x)
                         : "memory");
#else
            ((float*)Xs2)[flat] = x[goff >> 2];
#endif
        }
        #pragma unroll
        for (int j = 0; j < 8; ++j) {
            const int tl   = wt0 + j;
            const int t    = tb + tl;
            const int c    = t / kK;
            const int k    = t - c * kK;
            const int flat = buf * (2 * kWsN) + (tl >> 1) * (2 * kWsStr) + 2 * wo + (tl & 1);
            const uint32_t goff = (uint32_t)(((((wo * kCin + c) * kLout) + i) * kK + k) * 4);
#if USE_ASYNC_LDS
            asm volatile("global_load_async_to_lds_b32 %0, %1, %2"
                         :: "v"(wsBase + (uint32_t)flat * 4u), "v"(goff), "s"(w)
                         : "memory");
#else
            ((float*)Ws2)[flat] = w[goff >> 2];
#endif
        }
    };

    // ---- 8 K-steps of 4; each step: 2 A + 2 B b64 loads feed 4 WMMAs ----
    auto compute = [&](int buf) {
        const v2f* Xb = Xs2 + buf * kXsN;
        const v2f* Wb = Ws2 + buf * kWsN;
        #pragma unroll
        for (int s = 0; s < kKC / 4; ++s) {
            const int tp = 2 * s + half;
            const v2f a0 = Xb[tp * kXsStr + mg * 32 + l16];
            const v2f a1 = Xb[tp * kXsStr + mg * 32 + 16 + l16];
            const v2f b0 = Wb[tp * kWsStr + ng * 32 + l16];
            const v2f b1 = Wb[tp * kWsStr + ng * 32 + 16 + l16];
            acc[0][0] = __builtin_amdgcn_wmma_f32_16x16x4_f32(
                false, a0, false, b0, (short)0, acc[0][0], false, false);
            acc[0][1] = __builtin_amdgcn_wmma_f32_16x16x4_f32(
                false, a0, false, b1, (short)0, acc[0][1], false, false);
            acc[1][0] = __builtin_amdgcn_wmma_f32_16x16x4_f32(
                false, a1, false, b0, (short)0, acc[1][0], false, false);
            acc[1][1] = __builtin_amdgcn_wmma_f32_16x16x4_f32(
                false, a1, false, b1, (short)0, acc[1][1], false, false);
        }
    };

#if USE_ASYNC_LDS
    // In-order ASYNCcnt completion: waiting to <= kOpsPerFill retires the
    // older batch while the newest stays in flight.
    auto wait_prev = [&]() { asm volatile("s_wait_asynccnt %0" :: "i"(kOpsPerFill) : "memory"); };
    auto wait_all  = [&]() { asm volatile("s_wait_asynccnt 0" ::: "memory"); };
#else
    auto wait_prev = [&]() {};
    auto wait_all  = [&]() {};
#endif

    fill(0, 0);
    for (int ch = 0; ch < kChunks; ch += 2) {
        // buf0 holds chunk ch; prefetch ch+1 into buf1, then compute buf0.
        fill(ch + 1, 1);
        wait_prev();
        __syncthreads();
        compute(0);
        __syncthreads();            // buf0 free for refill after this
        // buf1 holds chunk ch+1; prefetch ch+2 into buf0 (if any).
        if (ch + 2 < kChunks) { fill(ch + 2, 0); wait_prev(); }
        else                  { wait_all(); }
        __syncthreads();
        compute(1);
        __syncthreads();            // buf1 free for refill after this
    }

    // ---- bias (reference adds C_IN * bias) + store ----
    // C/D layout: VGPR r -> M = r + 8*half ; N = l16
    #pragma unroll
    for (int ni = 0; ni < 2; ++ni) {
        const int   o  = (ng * 2 + ni) * 16 + l16;
        const float bv = (float)kCin * bias[(size_t)o * kLout + i];
        #pragma unroll
        for (int mi = 0; mi < 2; ++mi) {
            #pragma unroll
            for (int r = 0; r < 8; ++r) {
                const int brow = (mg * 2 + mi) * 16 + half * 8 + r;
                out[((size_t)brow * kCout + o) * kLout + i] = acc[mi][ni][r] + bv;
            }
        }
    }
}

extern "C" void kernel_launch(void* const* d_in, const int* in_sizes, int n_in,
                              void* d_out, int out_size, void* d_ws, size_t ws_size,
                              hipStream_t stream) {
    const float* x  = (const float*)d_in[0];
    const float* w  = (const float*)d_in[1];
    const float* b  = (const float*)d_in[2];
    float* out = (float*)d_out;

    dim3 grid(kLout);   // 1016 workgroups, one per output position
    dim3 block(256);    // 8 wave32s
    lc1d_wmma<<<grid, block, 0, stream>>>(x, w, b, out);
}